// mLSTMCell_24644522344828
// MI455X (gfx1250) — compile-verified
//
#include <hip/hip_runtime.h>
#include <cmath>

#define B_   64
#define IN_  1024
#define H_   1024
#define GN_EPS 1e-5f

typedef __attribute__((ext_vector_type(2))) float v2f;
typedef __attribute__((ext_vector_type(4))) float f32x4;
typedef __attribute__((ext_vector_type(8))) float v8f;

// ---------------------------------------------------------------------------
// Kernel 1: six gate GEMMs  out[g][b][h] = sum_k x[b][k] * W_g[h][k]
// One wave (32 threads) per 16x16 output tile. K staged via LDS in chunks of 8.
// Uses V_WMMA_F32_16X16X4_F32 (full fp32 precision; GEMM phase is ~1 us of
// traffic, far from the 512MB C-update bottleneck, so fp32 is the right call).
// A-frag (16x4 f32): lanes 0-15 M=0..15, v[0]=K0,v[1]=K1; lanes 16-31 K2/K3.
// B-frag (4x16 f32): lanes hold N=lane&15; v[0]=K0(lo)/K2(hi), v[1]=K1/K3.
// ---------------------------------------------------------------------------
__global__ __launch_bounds__(32) void gemm_gates(
    const float* __restrict__ x,
    const float* __restrict__ Wq, const float* __restrict__ Wk,
    const float* __restrict__ Wv, const float* __restrict__ Wi,
    const float* __restrict__ Wf, const float* __restrict__ Wo,
    float* __restrict__ gates)
{
    const int g = blockIdx.z;
    const float* W = (g == 0) ? Wq : (g == 1) ? Wk : (g == 2) ? Wv
                   : (g == 3) ? Wi : (g == 4) ? Wf : Wo;

    const int lane = threadIdx.x;            // 0..31
    const int h0 = blockIdx.x * 16;
    const int b0 = blockIdx.y * 16;

    __shared__ float lx[16 * 8];
    __shared__ float lw[16 * 8];

    const int row  = lane >> 1;              // 0..15 (load row)
    const int koff = (lane & 1) * 4;         // 0 or 4 (load col)
    const int fr   = lane & 15;              // frag row (A) / col (B)
    const int kh   = (lane >> 4) * 2;        // K sub-offset: 0 (lo half) / 2 (hi)

    v8f acc = {};

    for (int k0 = 0; k0 < IN_; k0 += 8) {
        f32x4 xa = *(const f32x4*)(x + (size_t)(b0 + row) * IN_ + k0 + koff);
        f32x4 wa = *(const f32x4*)(W + (size_t)(h0 + row) * IN_ + k0 + koff);
        __syncthreads();
        *(f32x4*)(lx + row * 8 + koff) = xa;
        *(f32x4*)(lw + row * 8 + koff) = wa;
        __syncthreads();
#pragma unroll
        for (int kk = 0; kk < 8; kk += 4) {
            const int ia = fr * 8 + kk + kh;
            v2f a = { lx[ia], lx[ia + 1] };
            v2f b = { lw[ia], lw[ia + 1] };
            acc = __builtin_amdgcn_wmma_f32_16x16x4_f32(
                      false, a, false, b, (short)0, acc, false, false);
        }
    }

    // C/D layout: VGPR j -> lanes 0-15: M=j; lanes 16-31: M=j+8; N=lane&15
    float* gout = gates + (size_t)g * B_ * H_;
#pragma unroll
    for (int j = 0; j < 8; ++j) {
        const int M = j + ((lane >> 4) << 3);
        gout[(size_t)(b0 + M) * H_ + h0 + fr] = acc[j];
    }
}

// ---------------------------------------------------------------------------
// Kernel 2: activations (in-place on gate slots 3/4/5), n, m, and
// denom[b] = max(sum_h n*q, 1).  One block per batch row.
// ---------------------------------------------------------------------------
__global__ __launch_bounds__(256) void act_nm_denom(
    float* __restrict__ gates,
    const float* __restrict__ bi, const float* __restrict__ bf,
    const float* __restrict__ bo,
    const float* __restrict__ n_prev, const float* __restrict__ m_prev,
    float* __restrict__ n_out, float* __restrict__ m_out,
    float* __restrict__ denom)
{
    const int b = blockIdx.x;
    const int t = threadIdx.x;
    const size_t base = (size_t)b * H_;
    const size_t BH = (size_t)B_ * H_;

    const float* qg = gates + 0 * BH;
    const float* kg = gates + 1 * BH;
    float* ig = gates + 3 * BH;
    float* fg = gates + 4 * BH;
    float* og = gates + 5 * BH;

    __shared__ float red[256];
    float acc = 0.f;

#pragma unroll
    for (int j = 0; j < H_ / 256; ++j) {
        const int h = t + j * 256;
        const float qv = qg[base + h];
        const float kv = kg[base + h];

        float iv = ig[base + h] + bi[h];
        iv = (iv > 20.f) ? iv : log1pf(expf(iv));           // softplus
        const float fv = 1.f / (1.f + expf(-(fg[base + h] + bf[h])));
        const float ov = 1.f / (1.f + expf(-(og[base + h] + bo[h])));
        ig[base + h] = iv;
        fg[base + h] = fv;
        og[base + h] = ov;

        const float nv = fv * n_prev[base + h] + iv * kv;
        n_out[base + h] = nv;
        m_out[base + h] = fmaxf(fv * m_prev[base + h], iv);
        acc += nv * qv;
    }

    red[t] = acc;
    __syncthreads();
    for (int s = 128; s > 0; s >>= 1) {
        if (t < s) red[t] += red[t + s];
        __syncthreads();
    }
    if (t == 0) denom[b] = fmaxf(red[0], 1.0f);
}

// ---------------------------------------------------------------------------
// Kernel 3: THE bandwidth kernel. C = f*C_prev + (i*k) (x) v, fused with
// h_raw[b,r] = sum_c C[b,r,c]*q[b,c].  512 MB of C traffic -> nontemporal
// 128-bit loads/stores (C >> L2, don't pollute). One wave per row, 8 rows
// (same b) per 256-thread block; q/v rows staged in LDS once per block.
// ---------------------------------------------------------------------------
__global__ __launch_bounds__(256) void c_update(
    const float* __restrict__ C_prev,
    const float* __restrict__ gates,
    float* __restrict__ C_out,
    float* __restrict__ h_raw)
{
    const int blk = blockIdx.x;          // B*H/8 blocks, 128 per batch row
    const int b   = blk >> 7;
    const int r0  = (blk & 127) * 8;
    const int t    = threadIdx.x;
    const int lane = t & 31;
    const int w    = t >> 5;

    const size_t BH = (size_t)B_ * H_;
    const float* qg = gates + 0 * BH + (size_t)b * H_;
    const float* kg = gates + 1 * BH + (size_t)b * H_;
    const float* vg = gates + 2 * BH + (size_t)b * H_;
    const float* ig = gates + 3 * BH + (size_t)b * H_;
    const float* fg = gates + 4 * BH + (size_t)b * H_;

    __shared__ f32x4 qs[H_ / 4];
    __shared__ f32x4 vs[H_ / 4];
    qs[t] = *(const f32x4*)(qg + t * 4);
    vs[t] = *(const f32x4*)(vg + t * 4);
    __syncthreads();

    const int r = r0 + w;
    const float frw = fg[r];
    const float ikr = ig[r] * kg[r];
    const size_t rowbase = ((size_t)b * H_ + r) * H_;
    const f32x4* cp = (const f32x4*)(C_prev + rowbase);
    f32x4*       co = (f32x4*)(C_out + rowbase);

    float hacc = 0.f;
#pragma unroll
    for (int it = 0; it < H_ / (32 * 4); ++it) {
        const int idx = lane + it * 32;
        f32x4 c_old = __builtin_nontemporal_load(cp + idx);
        f32x4 cn = frw * c_old + ikr * vs[idx];
        __builtin_nontemporal_store(cn, co + idx);
        const f32x4 qq = qs[idx];
        hacc += cn.x * qq.x + cn.y * qq.y + cn.z * qq.z + cn.w * qq.w;
    }

    // wave32 reduction
    for (int off = 16; off > 0; off >>= 1)
        hacc += __shfl_down(hacc, off, 32);
    if (lane == 0) h_raw[(size_t)b * H_ + r] = hacc;
}

// ---------------------------------------------------------------------------
// Kernel 4: h_tilde = h_raw/denom; GroupNorm(1,H) over the row; h = o*h_norm.
// Operates in place on d_out's h slot (block owns its row; all reads happen
// before the reduction barrier).
// ---------------------------------------------------------------------------
__global__ __launch_bounds__(256) void finalize_h(
    float* __restrict__ h_out,
    const float* __restrict__ gates,
    const float* __restrict__ denom,
    const float* __restrict__ gn_w, const float* __restrict__ gn_b)
{
    const int b = blockIdx.x;
    const int t = threadIdx.x;
    const size_t base = (size_t)b * H_;
    const float* og = gates + 5 * (size_t)B_ * H_ + base;

    __shared__ float s1[256];
    __shared__ float s2[256];

    const float inv_d = 1.f / denom[b];
    float vals[H_ / 256];
    float sum = 0.f, sq = 0.f;
#pragma unroll
    for (int j = 0; j < H_ / 256; ++j) {
        const int h = t + j * 256;
        const float ht = h_out[base + h] * inv_d;
        vals[j] = ht;
        sum += ht;
        sq  += ht * ht;
    }
    s1[t] = sum;
    s2[t] = sq;
    __syncthreads();
    for (int s = 128; s > 0; s >>= 1) {
        if (t < s) { s1[t] += s1[t + s]; s2[t] += s2[t + s]; }
        __syncthreads();
    }
    const float mu  = s1[0] * (1.f / H_);
    const float var = s2[0] * (1.f / H_) - mu * mu;
    const float rs  = rsqrtf(var + GN_EPS);

#pragma unroll
    for (int j = 0; j < H_ / 256; ++j) {
        const int h = t + j * 256;
        const float hn = (vals[j] - mu) * rs * gn_w[h] + gn_b[h];
        h_out[base + h] = og[h] * hn;
    }
}

// ---------------------------------------------------------------------------
extern "C" void kernel_launch(void* const* d_in, const int* in_sizes, int n_in,
                              void* d_out, int out_size, void* d_ws, size_t ws_size,
                              hipStream_t stream)
{
    const float* x      = (const float*)d_in[0];
    const float* C_prev = (const float*)d_in[1];
    const float* n_prev = (const float*)d_in[2];
    const float* m_prev = (const float*)d_in[3];
    const float* Wq = (const float*)d_in[4];
    const float* Wk = (const float*)d_in[5];
    const float* Wv = (const float*)d_in[6];
    const float* Wi = (const float*)d_in[7];
    const float* bi = (const float*)d_in[8];
    const float* Wf = (const float*)d_in[9];
    const float* bf = (const float*)d_in[10];
    const float* Wo = (const float*)d_in[11];
    const float* bo = (const float*)d_in[12];
    const float* gn_w = (const float*)d_in[13];
    const float* gn_b = (const float*)d_in[14];

    const size_t BH = (size_t)B_ * H_;
    float* out   = (float*)d_out;
    float* h_out = out;                       // [B,H]
    float* C_out = out + BH;                  // [B,H,H]
    float* n_out = C_out + BH * H_;           // [B,H]
    float* m_out = n_out + BH;                // [B,H]

    float* gates = (float*)d_ws;              // 6*[B,H]: q,k,v,i,f,o
    float* denom = gates + 6 * BH;            // [B]

    gemm_gates<<<dim3(H_ / 16, B_ / 16, 6), 32, 0, stream>>>(
        x, Wq, Wk, Wv, Wi, Wf, Wo, gates);
    act_nm_denom<<<B_, 256, 0, stream>>>(
        gates, bi, bf, bo, n_prev, m_prev, n_out, m_out, denom);
    c_update<<<(B_ * H_) / 8, 256, 0, stream>>>(
        C_prev, gates, C_out, h_out);
    finalize_h<<<B_, 256, 0, stream>>>(
        h_out, gates, denom, gn_w, gn_b);
}